// _RRoiAlign_45749991637436
// MI455X (gfx1250) — compile-verified
//
#include <hip/hip_runtime.h>
#include <math.h>

#define POOLED_H 8
#define POOLED_W 64

namespace {
constexpr int   kB   = 8;
constexpr int   kC   = 32;
constexpr int   kH   = 160;
constexpr int   kW   = 160;
constexpr int   kN   = 1024;
constexpr int   kHW  = kH * kW;          // 25600 floats per channel plane
constexpr float kScale = 0.25f;
}

// fdlibm __kernel_sin / __kernel_cos, valid for |x| <= pi/4 (our theta is
// in [-pi/4, pi/4] by construction: ang in [-45,45] degrees). Evaluated in
// double -> ~1 ulp(double) -> correctly-rounded float, matching/beating
// libm sinf/cosf without the Payne-Hanek slow path (~330 dead SALU instrs).
__device__ __forceinline__ void sincos_kernel(float t, float* s, float* c) {
    const double S1 = -1.66666666666666324348e-01;
    const double S2 =  8.33333333332248946124e-03;
    const double S3 = -1.98412698298579493134e-04;
    const double S4 =  2.75573137070700676789e-06;
    const double S5 = -2.50507602534068634195e-08;
    const double S6 =  1.58969099521155010221e-10;
    const double C1 =  4.16666666666666019037e-02;
    const double C2 = -1.38888888888741095749e-03;
    const double C3 =  2.48015872894767294178e-05;
    const double C4 = -2.75573143513906633035e-07;
    const double C5 =  2.08757232129817482790e-09;
    const double C6 = -1.13596475577881948265e-11;

    const double x = (double)t;
    const double z = x * x;
    const double sp = z * (S1 + z * (S2 + z * (S3 + z * (S4 + z * (S5 + z * S6)))));
    const double cp = z * z * (C1 + z * (C2 + z * (C3 + z * (C4 + z * (C5 + z * C6)))));
    *s = (float)(x + x * sp);
    *c = (float)(1.0 - 0.5 * z + cp);
}

// One thread = one (n, ph, pw) pooled bin across all 32 channels.
// Block: 256 threads = 4 pooled rows (ph) of ONE roi -> roi math is
// block-uniform and the compiler scalarizes it onto the SALU.
// tid & 63 = pw (coalesced 128B-per-wave32 stores), tid >> 6 = row-in-block.
__global__ __launch_bounds__(256)
void rroi_align_kernel(const float* __restrict__ feat,
                       const float* __restrict__ rois,
                       float* __restrict__ out)
{
    const int tid = threadIdx.x;
    const int pw  = tid & 63;
    const int n   = blockIdx.x >> 1;                          // 2 blocks per roi
    const int ph  = ((blockIdx.x & 1) << 2) | (tid >> 6);     // 0..7

    // ---- ROI parameters (uniform across block -> s_load) ----
    const float* __restrict__ r = rois + n * 6;
    const int   bidx = (int)r[0];
    const float cx = r[1], cy = r[2], hh = r[3], ww = r[4], ang = r[5];

    const float theta = ang * 0.017453292519943295f;          // pi/180
    float ca, sb;
    sincos_kernel(theta, &sb, &ca);

    const float Sx = ww * kScale * (1.0f / POOLED_W);
    const float Sy = hh * kScale * (1.0f / POOLED_H);
    const float dx = -0.5f * POOLED_W;                        // -32
    const float dy = -0.5f * POOLED_H;                        // -4
    const float Dx = cx * kScale;
    const float Dy = cy * kScale;

    const float M00 =  ca * Sx;
    const float M01 =  sb * Sy;
    const float M02 =  ca * Sx * dx + sb * Sy * dy + Dx;
    const float M10 = -sb * Sx;
    const float M11 =  ca * Sy;
    const float M12 = -sb * Sx * dx + ca * Sy * dy + Dy;

    // ---- 4 bin corners -> rounded, clipped bounding box ----
    const float pwf = (float)pw, phf = (float)ph;
    const float X0 = M00 * pwf        + M01 * phf        + M02;
    const float X1 = M00 * pwf        + M01 * (phf+1.f)  + M02;
    const float X2 = M00 * (pwf+1.f)  + M01 * phf        + M02;
    const float X3 = M00 * (pwf+1.f)  + M01 * (phf+1.f)  + M02;
    const float Y0 = M10 * pwf        + M11 * phf        + M12;
    const float Y1 = M10 * pwf        + M11 * (phf+1.f)  + M12;
    const float Y2 = M10 * (pwf+1.f)  + M11 * phf        + M12;
    const float Y3 = M10 * (pwf+1.f)  + M11 * (phf+1.f)  + M12;

    const float Xmin = fminf(fminf(X0, X1), fminf(X2, X3));
    const float Xmax = fmaxf(fmaxf(X0, X1), fmaxf(X2, X3));
    const float Ymin = fminf(fminf(Y0, Y1), fminf(Y2, Y3));
    const float Ymax = fmaxf(fmaxf(Y0, Y1), fmaxf(Y2, Y3));

    // reference _round(x) == floor(x + 0.5)
    const float left  = fmaxf(floorf(Xmin + 0.5f), 0.0f);
    const float right = fminf(floorf(Xmax + 0.5f), (float)(kW - 1));
    const float top   = fmaxf(floorf(Ymin + 0.5f), 0.0f);
    const float bot   = fminf(floorf(Ymax + 0.5f), (float)(kH - 1));

    const float bcx = (left + right) * 0.5f;
    const float bcy = (top  + bot )  * 0.5f;
    const float bl = floorf(bcx), br = ceilf(bcx);
    const float bt = floorf(bcy), bb = ceilf(bcy);
    const float rx = bcx - bl;
    const float ry = bcy - bt;

    // validity per corner axis (reference: yi > -1 && yi < H, etc.)
    const bool vxl = (bl > -1.0f) & (bl < (float)kW);
    const bool vxr = (br > -1.0f) & (br < (float)kW);
    const bool vyt = (bt > -1.0f) & (bt < (float)kH);
    const bool vyb = (bb > -1.0f) & (bb < (float)kH);

    const int xl = (int)fminf(fmaxf(bl, 0.0f), (float)(kW - 1));
    const int xr = (int)fminf(fmaxf(br, 0.0f), (float)(kW - 1));
    const int yt = (int)fminf(fmaxf(bt, 0.0f), (float)(kH - 1));
    const int yb = (int)fminf(fmaxf(bb, 0.0f), (float)(kH - 1));

    const int i_lt = yt * kW + xl;
    const int i_rt = yt * kW + xr;
    const int i_lb = yb * kW + xl;
    const int i_rb = yb * kW + xr;

    // fold validity into weights: where(valid, v, 0)*w == (valid ? w : 0)*v_clamped
    const float w_lt = (vyt && vxl) ? (1.0f - rx) * (1.0f - ry) : 0.0f;
    const float w_rt = (vyt && vxr) ? rx * (1.0f - ry)          : 0.0f;
    const float w_lb = (vyb && vxl) ? (1.0f - rx) * ry          : 0.0f;
    const float w_rb = (vyb && vxr) ? rx * ry                   : 0.0f;

    // ---- channel loop: 4 gathers + blend + NT store, stride = one plane ----
    const float* __restrict__ fp = feat + (size_t)bidx * (size_t)(kC * kHW);
    float* __restrict__ op = out
        + ((size_t)n * kC * (POOLED_H * POOLED_W))
        + (size_t)ph * POOLED_W + pw;

    #pragma unroll 8
    for (int c = 0; c < kC; ++c) {
        const float* __restrict__ fc = fp + (size_t)c * kHW;
        // Next unroll-group's lines live 8 planes (800KB) ahead: no HW
        // prefetcher catches that stride -> gfx1250 global_prefetch_b8.
        // Speculative prefetch: OOB past the last plane is silently dropped,
        // so no guard branch needed in the hot loop.
        __builtin_prefetch(fc + 8 * kHW + i_lt, 0, 0);
        __builtin_prefetch(fc + 8 * kHW + i_rb, 0, 0);

        const float v = w_lt * fc[i_lt] + w_rt * fc[i_rt]
                      + w_lb * fc[i_lb] + w_rb * fc[i_rb];
        // Output is 64MB write-once streaming data; NT store keeps the 25MB
        // feature gather working set resident in the 192MB L2.
        __builtin_nontemporal_store(v, op + (size_t)c * (POOLED_H * POOLED_W));
    }
}

extern "C" void kernel_launch(void* const* d_in, const int* in_sizes, int n_in,
                              void* d_out, int out_size, void* d_ws, size_t ws_size,
                              hipStream_t stream) {
    const float* feat = (const float*)d_in[0];   // (8, 32, 160, 160) f32
    const float* rois = (const float*)d_in[1];   // (1024, 6) f32
    float* out = (float*)d_out;                  // (1024, 32, 8, 64) f32

    const dim3 block(256);
    const dim3 grid(kN * POOLED_H / 4);          // 2048 blocks, 1 roi / 2 blocks
    hipLaunchKernelGGL(rroi_align_kernel, grid, block, 0, stream,
                       feat, rois, out);
}